// GraphDecoder_13211319403265
// MI455X (gfx1250) — compile-verified
//
#include <hip/hip_runtime.h>

#define NUM_ROWS   12288
#define CB_DIM     256
#define HID        512
#define NODES      16384
#define BLOCKS1    384
#define ROWS_PER_BLK (NUM_ROWS / BLOCKS1)   // 32

typedef __attribute__((ext_vector_type(2))) float v2f;
typedef __attribute__((ext_vector_type(8))) float v8f;

// ---------------------------------------------------------------------------
// Kernel 1: partial column sums of x [12288, 256].
// Each block streams 32 contiguous rows (32 KB) fully coalesced; thread t
// owns column t. Pure bandwidth-bound streaming — VALU adds keep up with
// 23.3 TB/s HBM trivially.
// ---------------------------------------------------------------------------
__global__ __launch_bounds__(256) void colsum_partial(const float* __restrict__ x,
                                                      float* __restrict__ part) {
  const int t = threadIdx.x;
  const int b = blockIdx.x;
  const float* p = x + (size_t)b * ROWS_PER_BLK * CB_DIM + t;
  float acc = 0.f;
#pragma unroll
  for (int r = 0; r < ROWS_PER_BLK; ++r) acc += p[(size_t)r * CB_DIM];
  part[b * CB_DIM + t] = acc;
}

// ---------------------------------------------------------------------------
// Kernel 2 (one 512-thread block = 16 waves):
//   1) deterministic reduce of partials -> m[256] = colmean(x) (divide by N)
//   2) v = m @ W1 via V_WMMA_F32_16X16X4_F32 (fp32-native WMMA):
//      A[16x4] holds m[k..k+3] broadcast to all 16 rows, so every row of the
//      16x16 D tile equals v[n0..n0+15]; accumulate over K=256 (64 WMMAs/tile)
//   3) hrow = relu(v + b1); dot = hrow . W2 via LDS tree reduction
//   4) broadcast relu(dot + b2) to all 16384 outputs
// ---------------------------------------------------------------------------
__global__ __launch_bounds__(512) void finalize(const float* __restrict__ part,
                                                const float* __restrict__ W1,
                                                const float* __restrict__ b1,
                                                const float* __restrict__ W2,
                                                const float* __restrict__ b2,
                                                float* __restrict__ out) {
  __shared__ float m_lds[CB_DIM];
  __shared__ float hrow[HID];
  __shared__ float red[512];

  const int t = threadIdx.x;

  // --- 1) final column reduction (fixed order -> deterministic) ---
  if (t < CB_DIM) {
    float acc = 0.f;
    for (int b = 0; b < BLOCKS1; ++b) acc += part[b * CB_DIM + t];
    m_lds[t] = acc * (1.0f / (float)NODES);
  }
  __syncthreads();

  // --- 2) v = m @ W1 with fp32 WMMA ---
  const int wave = t >> 5;
  const int lane = t & 31;
  const int hf   = lane >> 4;   // half-wave select (K+0/1 vs K+2/3 slots)
  const int nl   = lane & 15;

  for (int tile = wave; tile < HID / 16; tile += 16) {   // 32 N-tiles / 16 waves
    const int n0 = tile * 16;
    v8f c = {};
    for (int k0 = 0; k0 < CB_DIM; k0 += 4) {
      const int ka = k0 + 2 * hf;
      v2f a, bm;
      // A 16x4 layout: lanes 0-15 -> {K=k0,k0+1}, lanes 16-31 -> {K=k0+2,k0+3};
      // same value for every M row (broadcast trick).
      a.x = m_lds[ka];
      a.y = m_lds[ka + 1];
      // B 4x16 layout: lanes 0-15 -> rows k0,k0+1; lanes 16-31 -> rows k0+2,k0+3.
      bm.x = W1[(size_t)ka * HID + n0 + nl];
      bm.y = W1[(size_t)(ka + 1) * HID + n0 + nl];
      c = __builtin_amdgcn_wmma_f32_16x16x4_f32(
              /*neg_a=*/false, a, /*neg_b=*/false, bm,
              /*c_mod=*/(short)0, c, /*reuse_a=*/false, /*reuse_b=*/false);
    }
    // D row 0 lives in c[0], lanes 0-15 (N = lane). All D rows identical.
    if (lane < 16) {
      float v = c[0] + b1[n0 + nl];
      hrow[n0 + nl] = v > 0.f ? v : 0.f;
    }
  }
  __syncthreads();

  // --- 3) dot(hrow, W2) via deterministic LDS tree reduction ---
  red[t] = hrow[t] * W2[t];
  __syncthreads();
  for (int s = 256; s > 0; s >>= 1) {
    if (t < s) red[t] += red[t + s];
    __syncthreads();
  }

  // --- 4) broadcast scalar to all nodes ---
  float sc = red[0] + b2[0];
  sc = sc > 0.f ? sc : 0.f;
  for (int i = t; i < NODES; i += 512) out[i] = sc;
}

extern "C" void kernel_launch(void* const* d_in, const int* in_sizes, int n_in,
                              void* d_out, int out_size, void* d_ws, size_t ws_size,
                              hipStream_t stream) {
  const float* x  = (const float*)d_in[0];   // [12288, 256]
  const float* W1 = (const float*)d_in[1];   // [256, 512]
  const float* b1 = (const float*)d_in[2];   // [512]
  const float* W2 = (const float*)d_in[3];   // [512, 1]
  const float* b2 = (const float*)d_in[4];   // [1]
  float* out  = (float*)d_out;               // [16384]
  float* part = (float*)d_ws;                // [384 * 256] partial column sums

  colsum_partial<<<BLOCKS1, 256, 0, stream>>>(x, part);
  finalize<<<1, 512, 0, stream>>>(part, W1, b1, W2, b2, out);
}